// PolyMinLayer_63050119905525
// MI455X (gfx1250) — compile-verified
//
#include <hip/hip_runtime.h>

// PolyMin on MI455X (gfx1250):
// Latency-bound scalar Newton iteration -> single wave32, wave-uniform x.
// The dual dot-products p'(x), p''(x) per step are evaluated with the only
// fp32 WMMA shape, V_WMMA_F32_16X16X4_F32 (exact fp32 FMA, RNE), so the
// matrix unit does the polynomial evaluation at full reference precision.

typedef float v2f __attribute__((ext_vector_type(2)));
typedef float v8f __attribute__((ext_vector_type(8)));

#define PM_MAX_ITER   100
#define PM_GRAD_SQ_TOL 1e-12f

static __device__ __forceinline__ float pm_bcast(float v, int lane) {
    // uniform broadcast via v_readlane -> SGPR (keeps x / loop control scalar)
    return __int_as_float(__builtin_amdgcn_readlane(__float_as_int(v), lane));
}

// One WMMA evaluates g = p'(x) and h = p''(x).
//   A rows: 0 -> d1[0..3], 1 -> [d1[4],d1[5],0,0], 2 -> d2[0..3], 3 -> [d2[4],0,0,0]
//   B cols: 0 -> [1,x,x^2,x^3], 1 -> [x^4,x^5,0,0]
//   g = C[0][0] + C[1][1],  h = C[2][0] + C[3][1]
static __device__ __forceinline__ void pm_eval(float x, v2f a, float& g, float& h) {
    const int  L  = threadIdx.x & 31;
    const int  n  = L & 15;          // column index within half-wave
    const bool hi = (L >= 16);       // lanes 16-31 hold K=2 (VGPR0) / K=3 (VGPR1)

    const float x2 = x * x;
    const float x3 = x2 * x;
    const float x4 = x2 * x2;
    const float x5 = x4 * x;

    v2f b;
    if (!hi) {  // VGPR0 -> K=0 row, VGPR1 -> K=1 row
        b[0] = (n == 0) ? 1.0f : ((n == 1) ? x4 : 0.0f);
        b[1] = (n == 0) ? x    : ((n == 1) ? x5 : 0.0f);
    } else {    // VGPR0 -> K=2 row, VGPR1 -> K=3 row
        b[0] = (n == 0) ? x2 : 0.0f;
        b[1] = (n == 0) ? x3 : 0.0f;
    }

    v8f c = {0.f, 0.f, 0.f, 0.f, 0.f, 0.f, 0.f, 0.f};
    // 8 args: (neg_a, A, neg_b, B, c_mod, C, reuse_a, reuse_b)
    c = __builtin_amdgcn_wmma_f32_16x16x4_f32(false, a, false, b,
                                              (short)0, c, false, false);

    // C/D layout: VGPR m, lanes 0-15 -> row M=m, N=lane
    g = pm_bcast(c[0], 0) + pm_bcast(c[1], 1);   // C[0][0] + C[1][1]
    h = pm_bcast(c[2], 0) + pm_bcast(c[3], 1);   // C[2][0] + C[3][1]
}

__global__ void __launch_bounds__(32)
polymin_wmma_kernel(const float* __restrict__ poly,
                    const float* __restrict__ x_init,
                    float* __restrict__ out) {
    // Derivative coefficients (wave-uniform scalars; poly has 7 coeffs, low->high)
    const float d1_0 = poly[1] * 1.0f;
    const float d1_1 = poly[2] * 2.0f;
    const float d1_2 = poly[3] * 3.0f;
    const float d1_3 = poly[4] * 4.0f;
    const float d1_4 = poly[5] * 5.0f;
    const float d1_5 = poly[6] * 6.0f;
    const float d2_0 = d1_1 * 1.0f;
    const float d2_1 = d1_2 * 2.0f;
    const float d2_2 = d1_3 * 3.0f;
    const float d2_3 = d1_4 * 4.0f;
    const float d2_4 = d1_5 * 5.0f;

    // Pack the A matrix (loop-invariant). Lane L<16: M=L, K={0,1};
    // lane L>=16: M=L-16, K={2,3}.
    const int L = threadIdx.x & 31;
    float a0 = 0.0f, a1 = 0.0f;
    if      (L == 0)  { a0 = d1_0; a1 = d1_1; }  // row0 K0,K1
    else if (L == 16) { a0 = d1_2; a1 = d1_3; }  // row0 K2,K3
    else if (L == 1)  { a0 = d1_4; a1 = d1_5; }  // row1 K0,K1
    else if (L == 2)  { a0 = d2_0; a1 = d2_1; }  // row2 K0,K1
    else if (L == 18) { a0 = d2_2; a1 = d2_3; }  // row2 K2,K3
    else if (L == 3)  { a0 = d2_4; a1 = 0.0f; }  // row3 K0
    v2f a; a[0] = a0; a[1] = a1;

    float x = x_init[0];     // wave-uniform
    float g, h;
    pm_eval(x, a, g, h);     // g = p'(x0), h = p''(x0)

    // Reference loop: each body recomputes g=p'(x) (== previous g_new), h=p''(x),
    // steps, then checks p'(x_new)^2 <= tol. Single-eval-per-iteration form:
#pragma unroll 1
    for (int it = 0; it < PM_MAX_ITER; ++it) {
        const float step = (h > 0.0f) ? (g / h) : (0.1f * g);
        x = x - step;
        pm_eval(x, a, g, h);                 // g_new for the tol check, reused next iter
        if (g * g <= PM_GRAD_SQ_TOL) break;  // uniform scalar branch
    }

    if (threadIdx.x == 0) out[0] = x;
}

extern "C" void kernel_launch(void* const* d_in, const int* in_sizes, int n_in,
                              void* d_out, int out_size, void* d_ws, size_t ws_size,
                              hipStream_t stream) {
    (void)in_sizes; (void)n_in; (void)out_size; (void)d_ws; (void)ws_size;
    const float* poly   = (const float*)d_in[0];  // 7 floats, lowest degree first
    const float* x_init = (const float*)d_in[1];  // 1 float
    float*       out    = (float*)d_out;          // 1 float
    // One wave32 block: the problem is a serial scalar iteration (latency-bound);
    // EXEC stays all-1s so WMMA is legal throughout.
    polymin_wmma_kernel<<<1, 32, 0, stream>>>(poly, x_init, out);
}